// SNSCL_22531398435254
// MI455X (gfx1250) — compile-verified
//
#include <hip/hip_runtime.h>
#include <math.h>
#include <stdint.h>

typedef __attribute__((ext_vector_type(2))) float v2f;
typedef __attribute__((ext_vector_type(8))) float v8f;

#define TEMP_INV (1.0f / 0.07f)

// ---------------- CDNA5 async global->LDS helpers ----------------
// GLOBAL_LOAD_ASYNC_TO_LDS_B128 (opcode 98, tracked by ASYNCcnt).
// GV mode: 64-bit vaddr, SADDR = off. VDST VGPR carries the LDS byte address
// (low 32 bits of the flat shared address; aperture bits live in [63:32]).
__device__ __forceinline__ void async_ld_b128(uint32_t lds_off, const void* gptr) {
  asm volatile("global_load_async_to_lds_b128 %0, %1, off"
               :: "v"(lds_off), "v"((unsigned long long)(uintptr_t)gptr)
               : "memory");
}
__device__ __forceinline__ void wait_async0() {
  asm volatile("s_wait_asynccnt 0x0" ::: "memory");
}
__device__ __forceinline__ uint32_t lds_addr32(const void* p) {
  return (uint32_t)(uintptr_t)p;   // truncate flat addr -> LDS byte offset
}

// ---------------- reduction helpers (blockDim.x == 256) ----------------
__device__ __forceinline__ float block_reduce_sum_256(float v, float* s) {
  int tid = threadIdx.x;
  s[tid] = v;
  __syncthreads();
  for (int off = 128; off > 0; off >>= 1) {
    if (tid < off) s[tid] += s[tid + off];
    __syncthreads();
  }
  float r = s[0];
  __syncthreads();
  return r;
}

__device__ __forceinline__ float block_reduce_max_256(float v, float* s) {
  int tid = threadIdx.x;
  s[tid] = v;
  __syncthreads();
  for (int off = 128; off > 0; off >>= 1) {
    if (tid < off) s[tid] = fmaxf(s[tid], s[tid + off]);
    __syncthreads();
  }
  float r = s[0];
  __syncthreads();
  return r;
}

// ---------------- fp32 WMMA GEMM: C[M,N] = A[M,K] @ B[K,N] (+bias[N]) ----
// 256 threads = 8 wave32s; block tile 32(M) x 64(N); K in chunks of 32.
// NDIM/KDIM compile-time -> pure shift/immediate addressing.
// Double-buffered LDS fed by async b128 loads overlapped with the WMMA chain.
// All dims are exact multiples of the tile, so EXEC is all-ones (WMMA req).
template <int NDIM, int KDIM, bool HAS_BIAS>
__global__ __launch_bounds__(256) void gemm_wmma_f32(
    const float* __restrict__ A, const float* __restrict__ B,
    float* __restrict__ C, const float* __restrict__ bias) {
  // A stored row-major [m][k] with stride 36 (rows 144B => b128-aligned slots,
  // and 36*m mod 64 distinct across 16 lanes => conflict-free b64 frag reads).
  __shared__ float As[2][32][36];
  __shared__ float Bs[2][32][64];

  const int tid  = threadIdx.x;
  const int w    = tid >> 5;        // wave 0..7
  const int lane = tid & 31;
  const int half = lane >> 4;       // 0: lanes 0-15, 1: lanes 16-31
  const int ll   = lane & 15;
  const int wrow = (w >> 2) << 4;   // 0 / 16
  const int wcol = (w & 3) << 4;    // 0 / 16 / 32 / 48
  const int m0 = blockIdx.y << 5;
  const int n0 = blockIdx.x << 6;

  // Per-thread staging slots (chunk-invariant parts precomputed):
  //  A: 256 threads x 1 float4  -> 32x32 tile; m = tid>>3, k4 = (tid&7)*4
  //  B: 256 threads x 2 float4  -> 32x64 tile; k = idx>>4, n4 = (idx&15)*4
  const int am  = tid >> 3, ak4 = (tid & 7) << 2;
  const int bk0 = tid >> 4, bn4 = (tid & 15) << 2;
  const float* Ap  = A + (size_t)(m0 + am) * KDIM + ak4;
  const float* Bp0 = B + (size_t)bk0 * NDIM + (n0 + bn4);
  const float* Bp1 = B + (size_t)(bk0 + 16) * NDIM + (n0 + bn4);

  const uint32_t ldsA0 = lds_addr32(&As[0][am][ak4]);
  const uint32_t ldsB0 = lds_addr32(&Bs[0][bk0][bn4]);
  const uint32_t ldsB1 = lds_addr32(&Bs[0][bk0 + 16][bn4]);
  const uint32_t dA = (uint32_t)(sizeof(float) * 32 * 36);  // buf stride
  const uint32_t dB = (uint32_t)(sizeof(float) * 32 * 64);

  constexpr int NCH = KDIM / 32;

  // prologue: chunk 0 -> buffer 0
  async_ld_b128(ldsA0, Ap);
  async_ld_b128(ldsB0, Bp0);
  async_ld_b128(ldsB1, Bp1);
  wait_async0();
  __syncthreads();

  v8f acc = {0.f, 0.f, 0.f, 0.f, 0.f, 0.f, 0.f, 0.f};

#pragma unroll 1
  for (int c = 0; c < NCH; ++c) {
    const int buf = c & 1;
    if (c + 1 < NCH) {  // issue next chunk while this one computes
      const int nb = buf ^ 1;
      async_ld_b128(ldsA0 + nb * dA, Ap + (size_t)(c + 1) * 32);
      async_ld_b128(ldsB0 + nb * dB, Bp0 + (size_t)(c + 1) * 32 * NDIM);
      async_ld_b128(ldsB1 + nb * dB, Bp1 + (size_t)(c + 1) * 32 * NDIM);
    }
#pragma unroll
    for (int kk = 0; kk < 32; kk += 4) {
      const int kf = kk + 2 * half;  // A 16x4 frag: lanes 0-15 K=kk..kk+1, 16-31 K=kk+2..kk+3
      v2f a = *(const v2f*)&As[buf][wrow + ll][kf];   // ds_load_b64
      v2f b;
      b.x = Bs[buf][kf][wcol + ll];
      b.y = Bs[buf][kf + 1][wcol + ll];
      acc = __builtin_amdgcn_wmma_f32_16x16x4_f32(
          false, a, false, b, (short)0, acc, false, false);
    }
    if (c + 1 < NCH) wait_async0();  // this wave's asyncs landed
    __syncthreads();                 // everyone's landed / reads retired
  }

  // Epilogue: col is invariant across the 8 accumulator rows -> load bias
  // once per lane; stores become constant-immediate-offset global_store_b32.
  const int col = n0 + wcol + ll;
  const float bv = HAS_BIAS ? bias[col] : 0.0f;
  float* Cp = C + (size_t)(m0 + wrow + (half << 3)) * NDIM + col;
#pragma unroll
  for (int v = 0; v < 8; ++v) {  // C/D layout: VGPR v -> M=v (+8 for hi half)
    Cp[(size_t)v * NDIM] = acc[v] + bv;
  }
}

// ---------------- elementwise / small kernels ----------------
__global__ void ema_kernel(const float4* __restrict__ wk,
                           const float4* __restrict__ wq,
                           float4* __restrict__ out, int n4) {
  int i = blockIdx.x * 256 + threadIdx.x;
  if (i < n4) {
    float4 a = wk[i], b = wq[i], r;
    r.x = 0.999f * a.x + 0.001f * b.x;
    r.y = 0.999f * a.y + 0.001f * b.y;
    r.z = 0.999f * a.z + 0.001f * b.z;
    r.w = 0.999f * a.w + 0.001f * b.w;
    out[i] = r;
  }
}

// aug = eps + exp(0.5*lv)*mu ; partial[blk] = sum(1 + lv - mu^2 - exp(lv))
__global__ __launch_bounds__(256) void sampler_kernel(
    const float* __restrict__ mu, const float* __restrict__ lv,
    const float* __restrict__ eps, float* __restrict__ aug,
    float* __restrict__ partial) {
  __shared__ float s[256];
  float acc = 0.f;
  const int n = 128 * 2048;
  for (int i = blockIdx.x * 256 + threadIdx.x; i < n; i += gridDim.x * 256) {
    float m = mu[i], l = lv[i];
    aug[i] = eps[i] + __expf(0.5f * l) * m;
    acc += 1.0f + l - m * m - __expf(l);
  }
  float t = block_reduce_sum_256(acc, s);
  if (threadIdx.x == 0) partial[blockIdx.x] = t;
}

// in-place row L2 normalize, rows of length 256
__global__ __launch_bounds__(256) void l2norm_rows_kernel(float* __restrict__ X) {
  __shared__ float s[256];
  __shared__ float scale;
  int b = blockIdx.x, tid = threadIdx.x;
  float v = X[(size_t)b * 256 + tid];
  float ss = block_reduce_sum_256(v * v, s);
  if (tid == 0) scale = 1.0f / fmaxf(sqrtf(ss), 1e-12f);
  __syncthreads();
  X[(size_t)b * 256 + tid] = v * scale;
}

__global__ __launch_bounds__(256) void lpos_kernel(
    const float* __restrict__ q, const float* __restrict__ k,
    float* __restrict__ lpos) {
  __shared__ float s[256];
  int b = blockIdx.x, tid = threadIdx.x;
  float p = q[(size_t)b * 256 + tid] * k[(size_t)b * 256 + tid];
  float t = block_reduce_sum_256(p, s);
  if (tid == 0) lpos[b] = t;
}

// per-row max and log-sum-exp over {l_pos, dot3 row} in logits/T domain
__global__ __launch_bounds__(256) void rowstat_kernel(
    const float* __restrict__ dot3, const float* __restrict__ lpos,
    float* __restrict__ rowmax, float* __restrict__ rowls) {
  __shared__ float s[256];
  int b = blockIdx.x, tid = threadIdx.x;
  const float* row = dot3 + (size_t)b * 64000;
  float m = -3.4e38f;
  for (int n = tid; n < 64000; n += 256) m = fmaxf(m, row[n] * TEMP_INV);
  if (tid == 0) m = fmaxf(m, lpos[b] * TEMP_INV);
  float mv = block_reduce_max_256(m, s);
  float acc = 0.f;
  for (int n = tid; n < 64000; n += 256) acc += __expf(row[n] * TEMP_INV - mv);
  if (tid == 0) acc += __expf(lpos[b] * TEMP_INV - mv);
  float total = block_reduce_sum_256(acc, s);
  if (tid == 0) { rowmax[b] = mv; rowls[b] = __logf(total); }
}

// gather (pos-block-first permutation) + log_softmax write
__global__ __launch_bounds__(256) void write_logits_kernel(
    const float* __restrict__ dot3, const float* __restrict__ lpos,
    const int* __restrict__ labels, const float* __restrict__ rowmax,
    const float* __restrict__ rowls, float* __restrict__ out) {
  int col = blockIdx.x * 256 + threadIdx.x;
  int b = blockIdx.y;
  if (col >= 64001) return;
  int lab = labels[b & 127];  // labels2 = concat(labels, labels)
  float x;
  if (col == 0) {
    x = lpos[b];
  } else {
    int t = col - 1;
    int c, qq;
    if (t < 64) { c = lab; qq = t; }                  // positive class block
    else {                                            // negatives, order-preserving
      int t2 = t - 64;
      int jj = t2 >> 6; qq = t2 & 63;
      c = jj + (jj >= lab ? 1 : 0);
    }
    x = dot3[(size_t)b * 64000 + c * 64 + qq];
  }
  out[(size_t)b * 64001 + col] = x * TEMP_INV - rowmax[b] - rowls[b];
}

__global__ void fill_labels_kernel(float* __restrict__ out) {
  int col = blockIdx.x * 256 + threadIdx.x;
  int b = blockIdx.y;
  if (col >= 64001) return;
  out[(size_t)b * 64001 + col] = (col < 65) ? (1.0f / 65.0f) : 0.0f;
}

__global__ __launch_bounds__(256) void finalize_kl_kernel(
    const float* __restrict__ pq, const float* __restrict__ pk,
    float* __restrict__ out) {
  __shared__ float s[256];
  int tid = threadIdx.x;
  float tq = block_reduce_sum_256(pq[tid] + pq[tid + 256], s);
  float tk = block_reduce_sum_256(pk[tid] + pk[tid + 256], s);
  if (tid == 0) {
    float kl1 = -(0.5f / 128.0f) * tq;
    float kl2 = -(0.5f / 128.0f) * tk;
    *out = (kl1 + kl2) * 1.0e-4f;
  }
}

// ---------------- host orchestration ----------------
extern "C" void kernel_launch(void* const* d_in, const int* in_sizes, int n_in,
                              void* d_out, int out_size, void* d_ws, size_t ws_size,
                              hipStream_t stream) {
  (void)in_sizes; (void)n_in; (void)out_size; (void)ws_size;
  const float* im_q    = (const float*)d_in[0];
  const float* im_k    = (const float*)d_in[1];
  const int*   labels  = (const int*)  d_in[2];
  const float* eps_q   = (const float*)d_in[3];
  const float* eps_k   = (const float*)d_in[4];
  const float* Wq_enc  = (const float*)d_in[5];
  const float* Wq_proj = (const float*)d_in[6];
  const float* Wk_enc  = (const float*)d_in[7];
  const float* Wk_proj = (const float*)d_in[8];
  const float* W_mu    = (const float*)d_in[9];
  const float* b_mu    = (const float*)d_in[10];
  const float* W_var   = (const float*)d_in[11];
  const float* b_var   = (const float*)d_in[12];
  const float* queue   = (const float*)d_in[13];

  const int F = 2048, P = 256, B0 = 128, B = 256, CQ = 64000;
  const size_t LOG = (size_t)B * (CQ + 1);   // 16,384,256 floats per plane

  // workspace carve-up (~24.7 MB)
  float* ws = (float*)d_ws;
  size_t o = 0;
  float* wk_enc2  = ws + o; o += (size_t)F * F;
  float* wk_proj2 = ws + o; o += (size_t)F * P;
  float* k_f      = ws + o; o += (size_t)B0 * F;
  float* mu       = ws + o; o += (size_t)B0 * F;
  float* lv       = ws + o; o += (size_t)B0 * F;
  float* qf_aug   = ws + o; o += (size_t)B0 * F;
  float* kf_aug   = ws + o; o += (size_t)B0 * F;
  float* q_all    = ws + o; o += (size_t)B * P;
  float* k_all    = ws + o; o += (size_t)B * P;
  float* lpos     = ws + o; o += B;
  float* rowmax   = ws + o; o += B;
  float* rowls    = ws + o; o += B;
  float* part_q   = ws + o; o += 512;
  float* part_k   = ws + o; o += 512;

  float* out        = (float*)d_out;
  float* out_logits = out;                    // [256, 64001]
  float* out_labels = out + LOG;              // [256, 64001]
  float* out_qf     = out + 2 * LOG;          // [128, 2048]
  float* out_kl     = out + 2 * LOG + (size_t)B0 * F;  // scalar
  float* dot3       = out_labels;  // stage dot3 here; overwritten by labels after

  dim3 blk(256);

  // 1) EMA key-encoder weights (float4)
  ema_kernel<<<(F * F / 4 + 255) / 256, blk, 0, stream>>>(
      (const float4*)Wk_enc, (const float4*)Wq_enc, (float4*)wk_enc2, F * F / 4);
  ema_kernel<<<(F * P / 4 + 255) / 256, blk, 0, stream>>>(
      (const float4*)Wk_proj, (const float4*)Wq_proj, (float4*)wk_proj2, F * P / 4);

  // 2) encoders: q_f -> output slot directly; k_f -> ws
  gemm_wmma_f32<2048, 2048, false><<<dim3(F / 64, B0 / 32), blk, 0, stream>>>(im_q, Wq_enc, out_qf, nullptr);
  gemm_wmma_f32<2048, 2048, false><<<dim3(F / 64, B0 / 32), blk, 0, stream>>>(im_k, wk_enc2, k_f, nullptr);

  // 3) sampler (q): mu/log_var GEMMs + elementwise + KLD partials
  gemm_wmma_f32<2048, 2048, true><<<dim3(F / 64, B0 / 32), blk, 0, stream>>>(out_qf, W_mu, mu, b_mu);
  gemm_wmma_f32<2048, 2048, true><<<dim3(F / 64, B0 / 32), blk, 0, stream>>>(out_qf, W_var, lv, b_var);
  sampler_kernel<<<512, blk, 0, stream>>>(mu, lv, eps_q, qf_aug, part_q);

  // 4) sampler (k): reuse mu/lv buffers
  gemm_wmma_f32<2048, 2048, true><<<dim3(F / 64, B0 / 32), blk, 0, stream>>>(k_f, W_mu, mu, b_mu);
  gemm_wmma_f32<2048, 2048, true><<<dim3(F / 64, B0 / 32), blk, 0, stream>>>(k_f, W_var, lv, b_var);
  sampler_kernel<<<512, blk, 0, stream>>>(mu, lv, eps_k, kf_aug, part_k);

  // 5) projections -> q_all/k_all rows, then L2 norm + l_pos
  gemm_wmma_f32<256, 2048, false><<<dim3(P / 64, B0 / 32), blk, 0, stream>>>(out_qf, Wq_proj, q_all, nullptr);
  gemm_wmma_f32<256, 2048, false><<<dim3(P / 64, B0 / 32), blk, 0, stream>>>(qf_aug, Wq_proj, q_all + (size_t)B0 * P, nullptr);
  gemm_wmma_f32<256, 2048, false><<<dim3(P / 64, B0 / 32), blk, 0, stream>>>(k_f, wk_proj2, k_all, nullptr);
  gemm_wmma_f32<256, 2048, false><<<dim3(P / 64, B0 / 32), blk, 0, stream>>>(kf_aug, wk_proj2, k_all + (size_t)B0 * P, nullptr);
  l2norm_rows_kernel<<<B, blk, 0, stream>>>(q_all);
  l2norm_rows_kernel<<<B, blk, 0, stream>>>(k_all);
  lpos_kernel<<<B, blk, 0, stream>>>(q_all, k_all, lpos);

  // 6) big similarity GEMM: dot3 = q_all @ queue  [256 x 64000]
  gemm_wmma_f32<64000, 256, false><<<dim3(CQ / 64, B / 32), blk, 0, stream>>>(q_all, queue, dot3, nullptr);

  // 7) log-softmax stats, permuted logits write, then labels plane + KLD
  rowstat_kernel<<<B, blk, 0, stream>>>(dot3, lpos, rowmax, rowls);
  write_logits_kernel<<<dim3(251, B), blk, 0, stream>>>(dot3, lpos, labels, rowmax, rowls, out_logits);
  fill_labels_kernel<<<dim3(251, B), blk, 0, stream>>>(out_labels);
  finalize_kl_kernel<<<1, blk, 0, stream>>>(part_q, part_k, out_kl);
}